// EmorCRF_49460843381678
// MI455X (gfx1250) — compile-verified
//
#include <hip/hip_runtime.h>

// EMoR CRF apply, tuned for MI455X (gfx1250, wave32).
//
// Stage 1 (tiny): crf(K=1024 x 3, interleaved) = f0 + weight(3x11) @ basis(11x1024)
//                 via V_WMMA_F32_16X16X4_F32 (K padded 11->12, M padded 3->16).
// Stage 2 (bandwidth-bound): 50.3M-element stream, LUT lerp out of LDS.
//   Traffic = ~402 MB @ 23.3 TB/s => ~17 us floor; everything else must stay
//   off the memory path (LUT gathers are ds_load from a 12 KB LDS copy).

typedef __attribute__((ext_vector_type(2))) float v2f;
typedef __attribute__((ext_vector_type(8))) float v8f;

#define K_LUT 1024
#define EMOR_DIM 11

// ---------------------------------------------------------------------------
// Stage 1: CRF table via f32 WMMA. One block, 8 waves, 64 N-tiles of 16.
// A (16x12): lane m<16 holds {A[m][k0], A[m][k0+1]}, lanes 16-31 hold K+2.
// B (12x16): lanes 0-15 hold rows k0,k0+1 at col=lane; lanes 16-31 rows k0+2,k0+3.
// C/D (16x16): VGPR r = row r (lanes 0-15), row 8+r (lanes 16-31).
// ---------------------------------------------------------------------------
__global__ __launch_bounds__(256) void emor_crf_wmma(
    const float* __restrict__ weight,   // (3, 11) row-major
    const float* __restrict__ basis,    // (11, 1024) row-major
    const float* __restrict__ f0,       // (1024)
    float* __restrict__ crf)            // out: (1024, 3) interleaved
{
    const int lane = threadIdx.x & 31;
    const int wave = threadIdx.x >> 5;       // 0..7
    const int mrow = lane & 15;
    const int hi   = (lane >> 4) & 1;        // 1 for lanes 16..31

    // A fragments for the 3 k-steps (zero-padded beyond M=3 / K=11).
    v2f afrag[3];
#pragma unroll
    for (int ks = 0; ks < 3; ++ks) {
        const int kb = ks * 4 + hi * 2;
        afrag[ks].x = (mrow < 3 && kb     < EMOR_DIM) ? weight[mrow * EMOR_DIM + kb]     : 0.0f;
        afrag[ks].y = (mrow < 3 && kb + 1 < EMOR_DIM) ? weight[mrow * EMOR_DIM + kb + 1] : 0.0f;
    }

    // Each wave owns 8 of the 64 column tiles. All 32 lanes stay active
    // through the WMMAs (EXEC must be all 1s); only the store is predicated.
    for (int t = wave * 8; t < wave * 8 + 8; ++t) {
        const int n0 = t * 16;
        const int n  = n0 + mrow;
        const float f0v = f0[n];

        v8f c;
#pragma unroll
        for (int r = 0; r < 8; ++r) c[r] = f0v;   // crf = f0 + W@B : seed C with f0

#if __has_builtin(__builtin_amdgcn_wmma_f32_16x16x4_f32)
#pragma unroll
        for (int ks = 0; ks < 3; ++ks) {
            const int kb = ks * 4 + hi * 2;
            v2f b;
            b.x = (kb     < EMOR_DIM) ? basis[kb * K_LUT + n]       : 0.0f;
            b.y = (kb + 1 < EMOR_DIM) ? basis[(kb + 1) * K_LUT + n] : 0.0f;
            c = __builtin_amdgcn_wmma_f32_16x16x4_f32(
                    /*neg_a=*/false, afrag[ks],
                    /*neg_b=*/false, b,
                    /*c_mod=*/(short)0, c,
                    /*reuse_a=*/false, /*reuse_b=*/false);
        }
#else
        // Scalar fallback (only if the f32 WMMA builtin is absent).
        {
            float a0 = f0v, a1 = f0v, a2 = f0v;
            for (int d = 0; d < EMOR_DIM; ++d) {
                const float bv = basis[d * K_LUT + n];
                a0 = fmaf(weight[0 * EMOR_DIM + d], bv, a0);
                a1 = fmaf(weight[1 * EMOR_DIM + d], bv, a1);
                a2 = fmaf(weight[2 * EMOR_DIM + d], bv, a2);
            }
            c[0] = a0; c[1] = a1; c[2] = a2;
        }
#endif
        if (lane < 16) {                 // rows 0..2 live in c[0..2] of lanes 0..15
            crf[n * 3 + 0] = c[0];
            crf[n * 3 + 1] = c[1];
            crf[n * 3 + 2] = c[2];
        }
    }
}

// ---------------------------------------------------------------------------
// Stage 2: streaming LUT lerp. float4 loads/stores (global_load/store_b128),
// 12 KB LUT in LDS, grid-stride loop (~8 float4 per thread).
// Channel of flat element e is e % 3; base of a float4 at i is 4i, and
// 4i mod 3 == i mod 3, so one 32-bit mod per vector.
// ---------------------------------------------------------------------------
__global__ __launch_bounds__(256) void emor_apply(
    const float* __restrict__ hdr,
    const float* __restrict__ expo,     // 1 element
    const float* __restrict__ crf,      // (1024, 3) interleaved
    float* __restrict__ out,
    unsigned n)                         // total floats (= N_PIXELS*3)
{
    __shared__ float s_lut[K_LUT * 3];
    for (unsigned i = threadIdx.x; i < K_LUT * 3; i += blockDim.x)
        s_lut[i] = crf[i];
    __syncthreads();

    const float e = expo[0];
    const unsigned n4     = n >> 2;
    const unsigned stride = blockDim.x * gridDim.x;
    const float4* __restrict__ hdr4 = reinterpret_cast<const float4*>(hdr);
    float4* __restrict__ out4       = reinterpret_cast<float4*>(out);

    for (unsigned i = blockIdx.x * blockDim.x + threadIdx.x; i < n4; i += stride) {
        const float4 h = hdr4[i];
        int ch = (int)(i % 3u);
        float hv[4] = {h.x, h.y, h.z, h.w};
        float r[4];
#pragma unroll
        for (int j = 0; j < 4; ++j) {
            float u    = fminf(fmaxf(hv[j] * e, 0.0f), 1.0f);
            float pos  = u * (float)(K_LUT - 1);
            int   idx  = (int)pos;                       // pos >= 0 -> trunc == floor
            idx        = idx > (K_LUT - 2) ? (K_LUT - 2) : idx;
            float frac = pos - (float)idx;
            const int base = idx * 3 + ch;
            const float v0 = s_lut[base];
            const float v1 = s_lut[base + 3];
            r[j] = fmaf(frac, v1 - v0, v0);
            ch = (ch == 2) ? 0 : ch + 1;
        }
        out4[i] = make_float4(r[0], r[1], r[2], r[3]);
    }

    // Tail (n % 4) — zero for this problem size, kept for robustness.
    const unsigned tail = n & 3u;
    if (tail && blockIdx.x == 0 && threadIdx.x == 0) {
        for (unsigned k = n - tail; k < n; ++k) {
            float u    = fminf(fmaxf(hdr[k] * e, 0.0f), 1.0f);
            float pos  = u * (float)(K_LUT - 1);
            int   idx  = (int)pos;
            idx        = idx > (K_LUT - 2) ? (K_LUT - 2) : idx;
            float frac = pos - (float)idx;
            const int ch = (int)(k % 3u);
            const float v0 = s_lut[idx * 3 + ch];
            const float v1 = s_lut[idx * 3 + 3 + ch];
            out[k] = fmaf(frac, v1 - v0, v0);
        }
    }
}

extern "C" void kernel_launch(void* const* d_in, const int* in_sizes, int n_in,
                              void* d_out, int out_size, void* d_ws, size_t ws_size,
                              hipStream_t stream) {
    const float* hdr    = (const float*)d_in[0];   // (N_PIXELS, 3)
    const float* expo   = (const float*)d_in[1];   // (1,)
    const float* f0     = (const float*)d_in[2];   // (1, 1024)
    const float* basis  = (const float*)d_in[3];   // (11, 1024)
    const float* weight = (const float*)d_in[4];   // (3, 11)
    float* out = (float*)d_out;
    float* crf = (float*)d_ws;                     // 12 KB scratch: (1024,3)

    // Stage 1: one block (8 waves), fully f32 via WMMA.
    emor_crf_wmma<<<1, 256, 0, stream>>>(weight, basis, f0, crf);

    // Stage 2: bandwidth-bound stream.
    const unsigned n  = (unsigned)in_sizes[0];
    const unsigned n4 = n >> 2;
    const int threads = 256;
    const unsigned per_block = (unsigned)threads * 8u;   // ~8 float4 per thread
    unsigned blocks = (n4 + per_block - 1) / per_block;
    if (blocks < 1u) blocks = 1u;
    if (blocks > 16384u) blocks = 16384u;
    emor_apply<<<(int)blocks, threads, 0, stream>>>(hdr, expo, crf, out, n);
}